// SCM_Layer_61838939127827
// MI455X (gfx1250) — compile-verified
//
#include <hip/hip_runtime.h>
#include <hip/hip_bf16.h>
#include <math.h>

typedef float v2f __attribute__((ext_vector_type(2)));
typedef float v8f __attribute__((ext_vector_type(8)));

#define B_ROWS 4096
#define D_DIM  1024
#define K_DIM  64
#define H_DIM  64
#define DD_DIM 16
#define NIT    2
#define LN_EPS 1e-5f
#define NBINS  4096
#define GCAP   8192

// ---- output offsets (floats) ----
#define OFF_ZC   0
#define OFF_ZD   262144
#define OFF_ZMU  327680
#define OFF_ZLV  393216
#define OFF_A    458752
#define OFF_DAG  462848
#define OFF_HSIC 462849
#define OFF_KL   462850

__device__ __forceinline__ v8f wmma4(v2f a, v2f b, v8f c) {
  return __builtin_amdgcn_wmma_f32_16x16x4_f32(false, a, false, b, (short)0, c, false, false);
}
__device__ __forceinline__ float gelu_exact(float x) {
  return 0.5f * x * (1.0f + erff(x * 0.70710678118654752f));
}

// ---------- per-row mean / rstd of features ----------
__global__ void k_rowstats(const float* __restrict__ F, float* mean, float* rstd) {
  __shared__ float s1[256], s2[256];
  int b = blockIdx.x, t = threadIdx.x;
  float a = 0.f, q = 0.f;
  for (int d = t; d < D_DIM; d += 256) { float v = F[b*D_DIM + d]; a += v; q += v*v; }
  s1[t] = a; s2[t] = q; __syncthreads();
  for (int off = 128; off > 0; off >>= 1) {
    if (t < off) { s1[t] += s1[t+off]; s2[t] += s2[t+off]; }
    __syncthreads();
  }
  if (t == 0) {
    float m = s1[0] / (float)D_DIM;
    float v = s2[0] / (float)D_DIM - m*m;
    if (v < 0.f) v = 0.f;
    mean[b] = m; rstd[b] = rsqrtf(v + LN_EPS);
  }
}

// ---------- combined weight prep: Wc[d,c] = g[d]*W[d,c] ; cc[c] = b.W + bias ----------
__global__ void k_prepw(const float* ln_g, const float* dln_g,
                        const float* W_u, const float* W_db, float* Wc) {
  int idx = blockIdx.x * 256 + threadIdx.x;
  if (idx >= D_DIM * 128) return;
  int d = idx >> 7, c = idx & 127;
  Wc[idx] = (c < 64) ? ln_g[d] * W_u[d*64 + c] : dln_g[d] * W_db[d*64 + (c - 64)];
}
__global__ void k_prepc(const float* ln_b, const float* dln_b,
                        const float* W_u, const float* W_db,
                        const float* b_u, const float* b_db, float* cc) {
  __shared__ float s[256];
  int c = blockIdx.x, t = threadIdx.x;
  float a = 0.f;
  if (c < 64) { for (int d = t; d < D_DIM; d += 256) a += ln_b[d] * W_u[d*64 + c]; }
  else { int c2 = c - 64; for (int d = t; d < D_DIM; d += 256) a += dln_b[d] * W_db[d*64 + c2]; }
  s[t] = a; __syncthreads();
  for (int off = 128; off > 0; off >>= 1) { if (t < off) s[t] += s[t+off]; __syncthreads(); }
  if (t == 0) cc[c] = s[0] + (c < 64 ? b_u[c] : b_db[c - 64]);
}

// ---------- A = sigmoid(adj) * (1 - I) ----------
__global__ void k_adj(const float* logits, float* Aout) {
  int idx = blockIdx.x * 256 + threadIdx.x;
  if (idx >= K_DIM * K_DIM) return;
  int i = idx >> 6, j = idx & 63;
  float sg = 1.f / (1.f + expf(-logits[idx]));
  Aout[idx] = (i == j) ? 0.f : sg;
}

// ---------- WMMA GEMM: U[4096,128] = xn @ Wc + cc ; xn = (F - m)*rstd ----------
__global__ void k_gemm(const float* __restrict__ F, const float* __restrict__ mean,
                       const float* __restrict__ rstd, const float* __restrict__ Wc,
                       const float* __restrict__ cc, float* __restrict__ U) {
  int tile = blockIdx.x;           // 256 M-tiles x 8 N-tiles
  int tm = tile >> 3, tn = tile & 7;
  int lane = threadIdx.x;
  int h = lane >> 4, lr = lane & 15;
  int rowA = tm * 16 + lr;
  float m = mean[rowA], r = rstd[rowA];
  const float* fr = F + (size_t)rowA * D_DIM;
  int colB = tn * 16 + lr;
  v8f acc = {};
  for (int k0 = 0; k0 < D_DIM; k0 += 4) {
    int ka = k0 + 2 * h;
    v2f a, b;
    a.x = (fr[ka] - m) * r;
    a.y = (fr[ka + 1] - m) * r;
    b.x = Wc[ka * 128 + colB];
    b.y = Wc[(ka + 1) * 128 + colB];
    acc = wmma4(a, b, acc);
  }
  float cb = cc[colB];
  for (int i = 0; i < 8; ++i) {
    int gr = tm * 16 + i + 8 * h;   // C layout: vgpr i -> M=i (lanes0-15), M=i+8 (lanes16-31)
    U[(size_t)gr * 128 + colB] = acc[i] + cb;
  }
}

// ---------- causal structural loop: K tiny MLPs, one block per batch row ----------
__global__ void k_causal(const float* __restrict__ U, const float* __restrict__ Aadj,
                         const float* __restrict__ W1, const float* __restrict__ b1,
                         const float* __restrict__ W2, const float* __restrict__ b2,
                         float* __restrict__ zc) {
  __shared__ float As[K_DIM * K_DIM];
  __shared__ float zs[K_DIM];
  int b = blockIdx.x, k = threadIdx.x;
  for (int i = k; i < K_DIM * K_DIM; i += K_DIM) As[i] = Aadj[i];
  float u = U[(size_t)b * 128 + k];
  zs[k] = u;
  __syncthreads();
  float z = u;
  for (int it = 0; it < NIT; ++it) {
    float p = 0.f;
    for (int j = 0; j < K_DIM; ++j) p += zs[j] * As[j * 64 + k];
    const float* w1a = W1 + k * 128;        // W1[k,0,:]
    const float* w1b = W1 + k * 128 + 64;   // W1[k,1,:]
    const float* w2  = W2 + k * 64;
    const float* bb1 = b1 + k * 64;
    float acc = 0.f;
    for (int hh = 0; hh < H_DIM; ++hh) {
      float tpre = p * w1a[hh] + u * w1b[hh] + bb1[hh];
      acc += gelu_exact(tpre) * w2[hh];
    }
    z = acc + b2[k];
    __syncthreads();
    zs[k] = z;
    __syncthreads();
  }
  zc[(size_t)b * 64 + k] = z;
}

// ---------- VAE head: dh=gelu(dh_pre); mu/lv; z_d; per-row KL ----------
__global__ void k_domain(const float* __restrict__ U, const float* W_mu, const float* b_mu,
                         const float* W_lv, const float* b_lv, const float* __restrict__ eps,
                         float* zd, float* zmu, float* zlv, float* klrow) {
  __shared__ float dh[H_DIM];
  int b = blockIdx.x, t = threadIdx.x;
  dh[t] = gelu_exact(U[(size_t)b * 128 + 64 + t]);
  __syncthreads();
  if (t < DD_DIM) {
    float mu = b_mu[t], lv = b_lv[t];
    for (int hh = 0; hh < H_DIM; ++hh) {
      float d = dh[hh];
      mu += d * W_mu[hh * DD_DIM + t];
      lv += d * W_lv[hh * DD_DIM + t];
    }
    float z = mu + eps[(size_t)b * DD_DIM + t] * expf(0.5f * lv);
    zmu[(size_t)b * DD_DIM + t] = mu;
    zlv[(size_t)b * DD_DIM + t] = lv;
    zd[(size_t)b * DD_DIM + t] = z;
    float kt = 1.0f + lv - mu * mu - expf(lv);
    for (int off = 1; off < 16; off <<= 1) kt += __shfl_xor(kt, off, 32);
    if (t == 0) klrow[b] = kt;
  }
}

// ---------- row sum-of-squares ----------
__global__ void k_sumsq(const float* __restrict__ X, int KD, float* out) {
  int b = blockIdx.x, lane = threadIdx.x;
  float s = 0.f;
  for (int i = lane; i < KD; i += 32) { float v = X[(size_t)b * KD + i]; s += v * v; }
  for (int off = 1; off < 32; off <<= 1) s += __shfl_xor(s, off, 32);
  if (lane == 0) out[b] = s;
}

// ---------- 16x16 Gram tile via f32 WMMA ----------
__device__ __forceinline__ v8f gram16(const float* __restrict__ X, int KD,
                                      int i0, int j0, int h, int lr) {
  v8f G = {};
  const float* ar = X + (size_t)(i0 + lr) * KD;
  const float* br = X + (size_t)(j0 + lr) * KD;
  for (int k0 = 0; k0 < KD; k0 += 4) {
    v2f a, b;
    a.x = ar[k0 + 2*h]; a.y = ar[k0 + 2*h + 1];
    b.x = br[k0 + 2*h]; b.y = br[k0 + 2*h + 1];
    G = wmma4(a, b, G);
  }
  return G;
}

// ---------- Gram stats pass (median machinery): mode 0=max/count, 1=hist, 2=gather ----------
__global__ void k_gram_stats(const float* __restrict__ X, int KD, const float* __restrict__ xn,
                             unsigned* ctrl, unsigned* hist, float* gbuf, int mode) {
  int i0 = blockIdx.x * 16;
  int lane = threadIdx.x & 31, wave = threadIdx.x >> 5;
  int h = lane >> 4, lr = lane & 15;
  float lo = 0.f, hi = 1.f, scale = 0.f;
  if (mode >= 1) {
    lo = __uint_as_float(ctrl[2]); hi = __uint_as_float(ctrl[3]);
    scale = (float)NBINS / (hi - lo);
  }
  float xni[8]; int gi[8];
  for (int r = 0; r < 8; ++r) { gi[r] = i0 + r + 8*h; xni[r] = xn[gi[r]]; }
  float lmax = 0.f; unsigned lcnt = 0;
  for (int j0 = wave * 16; j0 < B_ROWS; j0 += 64) {
    v8f G = gram16(X, KD, i0, j0, h, lr);
    int gj = j0 + lr;
    float xnj = xn[gj];
    for (int r = 0; r < 8; ++r) {
      if (gi[r] >= gj) continue;                 // strict upper triangle
      float d2 = xni[r] + xnj - 2.f * G[r];
      if (d2 <= 0.f) continue;                   // positives only
      if (mode == 0) { lmax = fmaxf(lmax, d2); lcnt++; }
      else if (mode == 1) {
        if (d2 >= lo && d2 < hi) {
          int bin = (int)((d2 - lo) * scale);
          if (bin < 0) bin = 0; if (bin >= NBINS) bin = NBINS - 1;
          atomicAdd(&hist[bin], 1u);
        }
      } else {
        if (d2 >= lo && d2 < hi) {
          unsigned idx = atomicAdd(&ctrl[6], 1u);
          if (idx < GCAP) gbuf[idx] = d2;
        }
      }
    }
  }
  if (mode == 0) {
    atomicMax(&ctrl[0], __float_as_uint(lmax));
    if (lcnt) atomicAdd(&ctrl[1], lcnt);
  }
}

// ---------- select bin / set bounds ----------
__global__ void k_select(unsigned* ctrl, unsigned* hist, int level) {
  int t = threadIdx.x;
  if (level == 0) {
    if (t == 0) {
      unsigned cnt = ctrl[1];
      ctrl[5] = (cnt > 0) ? (cnt - 1) / 2 : 0;   // target rank
      float maxv = __uint_as_float(ctrl[0]);
      float hi = maxv * 1.0000005f + 1e-30f;
      ctrl[2] = __float_as_uint(0.f);
      ctrl[3] = __float_as_uint(hi);
      ctrl[4] = 0;
    }
  } else {
    if (t == 0) {
      unsigned target = ctrl[5], below = ctrl[4];
      unsigned tl = target - below;
      float lo = __uint_as_float(ctrl[2]), hi = __uint_as_float(ctrl[3]);
      float w = (hi - lo) / (float)NBINS;
      unsigned cum = 0; int bsel = NBINS - 1;
      for (int b = 0; b < NBINS; ++b) {
        unsigned c = hist[b];
        if (cum + c > tl) { bsel = b; break; }
        cum += c;
      }
      float nlo = lo + bsel * w;
      ctrl[2] = __float_as_uint(nlo);
      ctrl[3] = __float_as_uint(nlo + w);
      ctrl[4] = below + cum;
    }
    __syncthreads();
    for (int b = t; b < NBINS; b += blockDim.x) hist[b] = 0;
  }
}

// ---------- exact rank pick among gathered candidates -> gamma ----------
__global__ void k_pick(unsigned* ctrl, const float* gbuf, float* gammaOut) {
  __shared__ float meds;
  int t = threadIdx.x;
  if (t == 0) meds = -1.f;
  __syncthreads();
  unsigned cnt = ctrl[1];
  unsigned m = ctrl[6]; if (m > GCAP) m = GCAP;
  unsigned tl = ctrl[5] - ctrl[4];
  if (cnt > 0 && m > 0) {
    for (unsigned i = t; i < m; i += blockDim.x) {
      float v = gbuf[i];
      unsigned less = 0, eq = 0;
      for (unsigned j = 0; j < m; ++j) {
        float u = gbuf[j];
        less += (u < v) ? 1u : 0u;
        eq   += (u == v) ? 1u : 0u;
      }
      if (less <= tl && tl < less + eq) meds = v;   // value-based: deterministic
    }
  }
  __syncthreads();
  if (t == 0) {
    float med = meds;
    if (med < 0.f) med = __uint_as_float(ctrl[2]);  // fallback: bin lower bound
    float sigma = (cnt > 0) ? sqrtf(med) : 1.0f;
    sigma = fmaxf(sigma, 1e-6f);
    gammaOut[0] = 1.0f / (2.0f * sigma * sigma);
  }
}

// ---------- fused HSIC pass: kc/kd tiles, row sums, sum(kc*kd) ----------
__global__ void k_hsic(const float* __restrict__ Zc, const float* __restrict__ Zd,
                       const float* __restrict__ xnc, const float* __restrict__ xnd,
                       const float* gC, const float* gD,
                       float* rs_c, float* rs_d, float* prodP) {
  __shared__ float lrc[4][16], lrd[4][16], lp[4];
  int i0 = blockIdx.x * 16;
  int lane = threadIdx.x & 31, wave = threadIdx.x >> 5;
  int h = lane >> 4, lr = lane & 15;
  float gc = gC[0], gd = gD[0];
  float rc[8], rd[8], xnci[8], xndi[8];
  for (int r = 0; r < 8; ++r) {
    rc[r] = 0.f; rd[r] = 0.f;
    int gi = i0 + r + 8*h;
    xnci[r] = xnc[gi]; xndi[r] = xnd[gi];
  }
  float prod = 0.f;
  for (int j0 = wave * 16; j0 < B_ROWS; j0 += 64) {
    v8f Gc = gram16(Zc, 64, i0, j0, h, lr);
    v8f Gd = gram16(Zd, 16, i0, j0, h, lr);
    float xncj = xnc[j0 + lr], xndj = xnd[j0 + lr];
    for (int r = 0; r < 8; ++r) {
      float d2c = xnci[r] + xncj - 2.f * Gc[r]; if (d2c < 0.f) d2c = 0.f;
      float d2d = xndi[r] + xndj - 2.f * Gd[r]; if (d2d < 0.f) d2d = 0.f;
      float kc = expf(-gc * d2c);
      float kd = expf(-gd * d2d);
      prod += kc * kd; rc[r] += kc; rd[r] += kd;
    }
  }
  for (int r = 0; r < 8; ++r) {
    float vc = rc[r], vd = rd[r];
    for (int off = 1; off < 16; off <<= 1) { vc += __shfl_xor(vc, off, 32); vd += __shfl_xor(vd, off, 32); }
    if (lr == 0) { lrc[wave][r + 8*h] = vc; lrd[wave][r + 8*h] = vd; }
  }
  float p = prod;
  for (int off = 1; off < 32; off <<= 1) p += __shfl_xor(p, off, 32);
  if (lane == 0) lp[wave] = p;
  __syncthreads();
  if (threadIdx.x < 16) {
    int m = threadIdx.x;
    rs_c[i0 + m] = lrc[0][m] + lrc[1][m] + lrc[2][m] + lrc[3][m];
    rs_d[i0 + m] = lrd[0][m] + lrd[1][m] + lrd[2][m] + lrd[3][m];
  }
  if (threadIdx.x == 0) prodP[blockIdx.x] = lp[0] + lp[1] + lp[2] + lp[3];
}

// ---------- final scalar combine: hsic + domain KL ----------
__global__ void k_finalize(const float* prodP, const float* rs_c, const float* rs_d,
                           const float* klrow, float* out) {
  __shared__ float s0[256], s1[256], s2[256], s3[256], s4[256];
  int t = threadIdx.x;
  float p = prodP[t];   // 256 partials
  float sc = 0.f, sd = 0.f, scd = 0.f, kl = 0.f;
  for (int i = t; i < B_ROWS; i += 256) {
    float c = rs_c[i], d = rs_d[i];
    sc += c; sd += d; scd += c * d; kl += klrow[i];
  }
  s0[t] = p; s1[t] = sc; s2[t] = sd; s3[t] = scd; s4[t] = kl;
  __syncthreads();
  for (int off = 128; off > 0; off >>= 1) {
    if (t < off) { s0[t]+=s0[t+off]; s1[t]+=s1[t+off]; s2[t]+=s2[t+off]; s3[t]+=s3[t+off]; s4[t]+=s4[t+off]; }
    __syncthreads();
  }
  if (t == 0) {
    float nf = (float)B_ROWS;
    float hsic = (s0[0] - 2.0f * s3[0] / nf + s2[0] * s1[0] / (nf * nf)) / ((nf - 1.f) * (nf - 1.f));
    out[OFF_HSIC] = hsic;
    out[OFF_KL] = -0.5f * s4[0] / nf;
  }
}

// ---------- trace(expm(A*A)) - K via scaling-and-squaring in LDS ----------
__global__ void k_expm(const float* __restrict__ Aadj, float* out) {
  __shared__ float X[4096], R[4096], S[4096];
  __shared__ float red[64];
  __shared__ int ssh;
  int t = threadIdx.x;
  for (int i = t; i < 4096; i += 256) { float a = Aadj[i]; X[i] = a * a; }
  __syncthreads();
  if (t < 64) { float cs = 0.f; for (int i = 0; i < 64; ++i) cs += X[i*64 + t]; red[t] = cs; }
  __syncthreads();
  if (t == 0) {
    float nrm = 0.f;
    for (int j = 0; j < 64; ++j) nrm = fmaxf(nrm, red[j]);
    int s = 0;
    while (nrm > 0.25f && s < 32) { nrm *= 0.5f; s++; }
    ssh = s;
  }
  __syncthreads();
  int s = ssh;
  float scl = ldexpf(1.0f, -s);
  for (int i = t; i < 4096; i += 256) X[i] *= scl;
  for (int i = t; i < 4096; i += 256) R[i] = ((i >> 6) == (i & 63)) ? 1.f : 0.f;
  __syncthreads();
  // Horner Taylor: R = I + X/k * R, k = 24..1
  for (int k = 24; k >= 1; --k) {
    float inv = 1.0f / (float)k;
    for (int e = t; e < 4096; e += 256) {
      int r = e >> 6, c = e & 63;
      float acc = 0.f;
      for (int q = 0; q < 64; ++q) acc += X[r*64 + q] * R[q*64 + c];
      S[e] = acc * inv + ((r == c) ? 1.f : 0.f);
    }
    __syncthreads();
    for (int e = t; e < 4096; e += 256) R[e] = S[e];
    __syncthreads();
  }
  for (int sq = 0; sq < s; ++sq) {
    for (int e = t; e < 4096; e += 256) {
      int r = e >> 6, c = e & 63;
      float acc = 0.f;
      for (int q = 0; q < 64; ++q) acc += R[r*64 + q] * R[q*64 + c];
      S[e] = acc;
    }
    __syncthreads();
    for (int e = t; e < 4096; e += 256) R[e] = S[e];
    __syncthreads();
  }
  if (t == 0) {
    float tr = 0.f;
    for (int i = 0; i < 64; ++i) tr += R[i*64 + i];
    out[OFF_DAG] = tr - (float)K_DIM;
  }
}

// ================= host launch =================
extern "C" void kernel_launch(void* const* d_in, const int* in_sizes, int n_in,
                              void* d_out, int out_size, void* d_ws, size_t ws_size,
                              hipStream_t stream) {
  (void)in_sizes; (void)n_in; (void)out_size; (void)ws_size;
  const float* F       = (const float*)d_in[0];
  const float* ln_g    = (const float*)d_in[1];
  const float* ln_b    = (const float*)d_in[2];
  const float* W_u     = (const float*)d_in[3];
  const float* b_u     = (const float*)d_in[4];
  const float* W1      = (const float*)d_in[5];
  const float* b1      = (const float*)d_in[6];
  const float* W2      = (const float*)d_in[7];
  const float* b2      = (const float*)d_in[8];
  const float* adj     = (const float*)d_in[9];
  const float* dln_g   = (const float*)d_in[10];
  const float* dln_b   = (const float*)d_in[11];
  const float* W_db    = (const float*)d_in[12];
  const float* b_db    = (const float*)d_in[13];
  const float* W_mu    = (const float*)d_in[14];
  const float* b_mu    = (const float*)d_in[15];
  const float* W_lv    = (const float*)d_in[16];
  const float* b_lv    = (const float*)d_in[17];
  const float* eps     = (const float*)d_in[18];
  float* out = (float*)d_out;

  // ---- workspace layout (floats) ----
  float* ws = (float*)d_ws;
  float* mean   = ws;                    // 4096
  float* rstd   = ws + 4096;             // 4096
  float* Wc     = ws + 8192;             // 131072
  float* cc     = ws + 139264;           // 128
  float* U      = ws + 139392;           // 524288
  float* xnc    = ws + 663680;           // 4096
  float* xnd    = ws + 667776;           // 4096
  float* rs_c   = ws + 671872;           // 4096
  float* rs_d   = ws + 675968;           // 4096
  float* prodP  = ws + 680064;           // 256
  float* klrow  = ws + 680320;           // 4096
  unsigned* ctrl = (unsigned*)(ws + 684416); // 16
  unsigned* hist = (unsigned*)(ws + 684432); // 4096
  float* gbuf   = ws + 688528;           // 8192
  float* gammaC = ws + 696720;           // 1
  float* gammaD = ws + 696721;           // 1

  float* Aout = out + OFF_A;
  float* zc   = out + OFF_ZC;
  float* zd   = out + OFF_ZD;
  float* zmu  = out + OFF_ZMU;
  float* zlv  = out + OFF_ZLV;

  // ---- stage 1: stats, weights, adjacency ----
  k_rowstats<<<B_ROWS, 256, 0, stream>>>(F, mean, rstd);
  k_prepw<<<(D_DIM*128 + 255)/256, 256, 0, stream>>>(ln_g, dln_g, W_u, W_db, Wc);
  k_prepc<<<128, 256, 0, stream>>>(ln_b, dln_b, W_u, W_db, b_u, b_db, cc);
  k_adj<<<(K_DIM*K_DIM + 255)/256, 256, 0, stream>>>(adj, Aout);

  // ---- stage 2: fused layernorm+projection GEMM (WMMA f32) ----
  k_gemm<<<2048, 32, 0, stream>>>(F, mean, rstd, Wc, cc, U);

  // ---- stage 3: causal loop + VAE head ----
  k_causal<<<B_ROWS, 64, 0, stream>>>(U, Aout, W1, b1, W2, b2, zc);
  k_domain<<<B_ROWS, 64, 0, stream>>>(U, W_mu, b_mu, W_lv, b_lv, eps, zd, zmu, zlv, klrow);

  // ---- stage 4: row norms ----
  k_sumsq<<<B_ROWS, 32, 0, stream>>>(zc, 64, xnc);
  k_sumsq<<<B_ROWS, 32, 0, stream>>>(zd, 16, xnd);

  // ---- stage 5: median heuristic (histogram refinement, WMMA Gram recompute) ----
  // kc path
  hipMemsetAsync(ctrl, 0, (16 + NBINS) * sizeof(unsigned), stream);
  k_gram_stats<<<256, 128, 0, stream>>>(zc, 64, xnc, ctrl, hist, gbuf, 0);
  k_select<<<1, 256, 0, stream>>>(ctrl, hist, 0);
  k_gram_stats<<<256, 128, 0, stream>>>(zc, 64, xnc, ctrl, hist, gbuf, 1);
  k_select<<<1, 256, 0, stream>>>(ctrl, hist, 1);
  k_gram_stats<<<256, 128, 0, stream>>>(zc, 64, xnc, ctrl, hist, gbuf, 1);
  k_select<<<1, 256, 0, stream>>>(ctrl, hist, 1);
  k_gram_stats<<<256, 128, 0, stream>>>(zc, 64, xnc, ctrl, hist, gbuf, 2);
  k_pick<<<1, 256, 0, stream>>>(ctrl, gbuf, gammaC);
  // kd path
  hipMemsetAsync(ctrl, 0, (16 + NBINS) * sizeof(unsigned), stream);
  k_gram_stats<<<256, 128, 0, stream>>>(zd, 16, xnd, ctrl, hist, gbuf, 0);
  k_select<<<1, 256, 0, stream>>>(ctrl, hist, 0);
  k_gram_stats<<<256, 128, 0, stream>>>(zd, 16, xnd, ctrl, hist, gbuf, 1);
  k_select<<<1, 256, 0, stream>>>(ctrl, hist, 1);
  k_gram_stats<<<256, 128, 0, stream>>>(zd, 16, xnd, ctrl, hist, gbuf, 1);
  k_select<<<1, 256, 0, stream>>>(ctrl, hist, 1);
  k_gram_stats<<<256, 128, 0, stream>>>(zd, 16, xnd, ctrl, hist, gbuf, 2);
  k_pick<<<1, 256, 0, stream>>>(ctrl, gbuf, gammaD);

  // ---- stage 6: fused HSIC + scalar losses ----
  k_hsic<<<256, 128, 0, stream>>>(zc, zd, xnc, xnd, gammaC, gammaD, rs_c, rs_d, prodP);
  k_finalize<<<1, 256, 0, stream>>>(prodP, rs_c, rs_d, klrow, out);
  k_expm<<<1, 256, 0, stream>>>(Aout, out);
}